// StructuredSpinnerLayer_25778393711187
// MI455X (gfx1250) — compile-verified
//
#include <hip/hip_runtime.h>

typedef __attribute__((ext_vector_type(2))) float v2f;
typedef __attribute__((ext_vector_type(8))) float v8f;

#define DDIM 2048
#define TT   16          // t-columns per block tile
#define LROW 17          // padded LDS row stride (floats)
#define NTHREADS 256

__device__ __forceinline__ float hsign(int i, int j) {
  return (__builtin_popcount(i & j) & 1) ? -1.0f : 1.0f;
}

__device__ __forceinline__ void fwht8(float x[8]) {
  float y[8];
  y[0]=x[0]+x[1]; y[1]=x[0]-x[1]; y[2]=x[2]+x[3]; y[3]=x[2]-x[3];
  y[4]=x[4]+x[5]; y[5]=x[4]-x[5]; y[6]=x[6]+x[7]; y[7]=x[6]-x[7];
  x[0]=y[0]+y[2]; x[2]=y[0]-y[2]; x[1]=y[1]+y[3]; x[3]=y[1]-y[3];
  x[4]=y[4]+y[6]; x[6]=y[4]-y[6]; x[5]=y[5]+y[7]; x[7]=y[5]-y[7];
  y[0]=x[0]+x[4]; y[4]=x[0]-x[4]; y[1]=x[1]+x[5]; y[5]=x[1]-x[5];
  y[2]=x[2]+x[6]; y[6]=x[2]-x[6]; y[3]=x[3]+x[7]; y[7]=x[3]-x[7];
#pragma unroll
  for (int j = 0; j < 8; ++j) x[j] = y[j];
}

// W = H16 @ M on rows {rowbase + stride*j, j=0..15}, cols 0..15 of the LDS
// tile, via 4 chained v_wmma_f32_16x16x4_f32 (K-chunks of 4). In-place.
__device__ __forceinline__ void h16_product(float* __restrict__ tile,
                                            const v2f A[4],
                                            int lane, int rowbase, int stride) {
  const int n    = lane & 15;          // output / B column
  const int koff = (lane >> 4) << 1;   // K offset: 0 (lanes 0-15) or 2 (16-31)
  v8f c = {0.f, 0.f, 0.f, 0.f, 0.f, 0.f, 0.f, 0.f};
#pragma unroll
  for (int k = 0; k < 4; ++k) {
    v2f bb;  // B 4x16: v0 = rows K/K+2 split across half-waves; v1 = K+1/K+3
    bb.x = tile[(rowbase + stride * (4 * k + koff))     * LROW + n];
    bb.y = tile[(rowbase + stride * (4 * k + koff + 1)) * LROW + n];
    c = __builtin_amdgcn_wmma_f32_16x16x4_f32(false, A[k], false, bb,
                                              (short)0, c, false, false);
  }
  const int rhalf = (lane >> 4) << 3;  // C/D rows v (lanes 0-15) / v+8 (16-31)
#pragma unroll
  for (int v = 0; v < 8; ++v)
    tile[(rowbase + stride * (v + rhalf)) * LROW + n] = c[v];
}

__global__ void __launch_bounds__(NTHREADS)
spinner_main(const float* __restrict__ z,
             const float* __restrict__ d1v,
             const float* __restrict__ d2v,
             const float* __restrict__ d3v,
             const float* __restrict__ bv,
             float* __restrict__ out, int T) {
  extern __shared__ float tile[];          // DDIM * LROW floats
  const int tid  = threadIdx.x;
  const int lane = tid & 31;
  const int wave = tid >> 5;

  const int blocksPerSlab = T / TT;
  const int slab = blockIdx.x / blocksPerSlab;        // k*M + m
  const int t0   = (blockIdx.x % blocksPerSlab) * TT;
  const size_t gbase = (size_t)slab * DDIM * T + t0;

  const float s = 0.022097086912079608f;   // 1/sqrt(2048)

  // Constant A operand: columns of H16 per the 16x4 f32 A layout.
  v2f A[4];
  {
    const int m = lane & 15;
    const int koff = (lane >> 4) << 1;
#pragma unroll
    for (int k = 0; k < 4; ++k) {
      A[k].x = hsign(m, 4 * k + koff);
      A[k].y = hsign(m, 4 * k + koff + 1);
    }
  }

  // Stage in LDS (d1-scaled), b128 global loads: 4 consecutive t per thread.
  for (int e4 = tid; e4 < (DDIM * TT) / 4; e4 += NTHREADS) {
    int d = e4 >> 2;
    int c = (e4 & 3) << 2;
    float4 v = *(const float4*)(z + gbase + (size_t)d * T + c);
    float sc = d1v[d];
    float* p = tile + d * LROW + c;
    p[0] = v.x * sc; p[1] = v.y * sc; p[2] = v.z * sc; p[3] = v.w * sc;
  }
  __syncthreads();

  for (int rep = 0; rep < 3; ++rep) {
    // Pass 1: H16 over d-bits 0-3 (row stride 1).
#pragma unroll 1
    for (int i = 0; i < 16; ++i)
      h16_product(tile, A, lane, (wave * 16 + i) * 16, 1);
    __syncthreads();

    // Pass 2: H16 over d-bits 4-7 (row stride 16).
#pragma unroll 1
    for (int i = 0; i < 16; ++i) {
      int g = wave * 16 + i;                       // g = h*16 + l
      h16_product(tile, A, lane, ((g >> 4) << 8) + (g & 15), 16);
    }
    __syncthreads();

    if (rep < 2) {
      // Pass 3: H8 over d-bits 8-10 (stride 256), fused diagonal scaling.
      const float* dv = (rep == 0) ? d2v : d3v;
#pragma unroll 1
      for (int q = 0; q < 16; ++q) {
        int combo = tid + (q << 8);
        int t = combo & 15;
        int l = combo >> 4;                        // 0..255 = d-bits 0-7
        float x[8];
#pragma unroll
        for (int j = 0; j < 8; ++j) x[j] = tile[(l + (j << 8)) * LROW + t];
        fwht8(x);
#pragma unroll
        for (int j = 0; j < 8; ++j) {
          int row = l + (j << 8);
          tile[row * LROW + t] = x[j] * (dv[row] * s);
        }
      }
      __syncthreads();
    } else {
      // Final pass 3: H8 + scale + bias, b128 global stores
      // (each thread owns 8 rows x 4 consecutive t).
#pragma unroll 1
      for (int q = 0; q < 4; ++q) {
        int combo = tid + (q << 8);                // 0..1023
        int tbase = (combo & 3) << 2;              // 0,4,8,12
        int l     = combo >> 2;                    // 0..255
        float y[8][4];
#pragma unroll
        for (int c = 0; c < 4; ++c) {
          float x[8];
#pragma unroll
          for (int j = 0; j < 8; ++j)
            x[j] = tile[(l + (j << 8)) * LROW + tbase + c];
          fwht8(x);
#pragma unroll
          for (int j = 0; j < 8; ++j) y[j][c] = x[j];
        }
#pragma unroll
        for (int j = 0; j < 8; ++j) {
          int row = l + (j << 8);
          float bb = bv[row];
          float4 o;
          o.x = y[j][0] * s + bb;
          o.y = y[j][1] * s + bb;
          o.z = y[j][2] * s + bb;
          o.w = y[j][3] * s + bb;
          *(float4*)(out + gbase + (size_t)row * T + tbase) = o;
        }
      }
    }
  }
}

__global__ void __launch_bounds__(NTHREADS)
spinner_sldj(const float* __restrict__ d1v, const float* __restrict__ d2v,
             const float* __restrict__ d3v, const float* __restrict__ sldj_in,
             float* __restrict__ sldj_out, int D, int KM) {
  __shared__ float red[NTHREADS];
  const int tid = threadIdx.x;
  float acc = 0.f;
  for (int d = tid; d < D; d += NTHREADS)
    acc += __logf(fabsf(d1v[d])) + __logf(fabsf(d2v[d])) + __logf(fabsf(d3v[d]));
  red[tid] = acc;
  __syncthreads();
  for (int off = NTHREADS / 2; off > 0; off >>= 1) {
    if (tid < off) red[tid] += red[tid + off];
    __syncthreads();
  }
  const float ld = red[0];
  for (int i = tid; i < KM; i += NTHREADS)
    sldj_out[i] = sldj_in[i] + ld;
}

extern "C" void kernel_launch(void* const* d_in, const int* in_sizes, int n_in,
                              void* d_out, int out_size, void* d_ws, size_t ws_size,
                              hipStream_t stream) {
  const float* z    = (const float*)d_in[0];
  const float* d1   = (const float*)d_in[1];
  const float* d2   = (const float*)d_in[2];
  const float* d3   = (const float*)d_in[3];
  const float* b    = (const float*)d_in[4];
  const float* sldj = (const float*)d_in[5];

  const int zsize = in_sizes[0];           // K*M*D*T
  const int D     = in_sizes[1];           // 2048
  const int KM    = in_sizes[5];           // 128
  const int T     = zsize / (KM * D);      // 128

  float* zout    = (float*)d_out;
  float* sldjout = zout + (size_t)zsize;

  const size_t ldsBytes = (size_t)DDIM * LROW * sizeof(float);  // ~136 KB
  (void)hipFuncSetAttribute((const void*)spinner_main,
                            hipFuncAttributeMaxDynamicSharedMemorySize,
                            (int)ldsBytes);

  dim3 grid(KM * (T / TT));                // 1024 blocks
  spinner_main<<<grid, NTHREADS, ldsBytes, stream>>>(z, d1, d2, d3, b, zout, T);
  spinner_sldj<<<1, NTHREADS, 0, stream>>>(d1, d2, d3, sldj, sldjout, D, KM);
}